// AssociatorLoss_20564303414105
// MI455X (gfx1250) — compile-verified
//
#include <hip/hip_runtime.h>

typedef __attribute__((ext_vector_type(2))) float v2f;
typedef __attribute__((ext_vector_type(8))) float v8f;

#define LN2F 0.69314718055994530942f

// Padded LDS layout for the "one" slice: one[l, j, k] at l*S_L + j*S_J + k.
// S_J = 36  -> half-wave offset 8*S_J = 288 = 32 (mod 64 banks)
// S_L = 1154 (>= 31*S_J + 32, and = 2 mod 64) -> lane offset l*S_L = 2*l (mod 64)
// => phase-2 gather: lanes 0-15 hit banks {c+2t}, lanes 16-31 {c+32+2t}: all 32 distinct.
#define S_J 36
#define S_L 1154
#define LDS_BYTES (32 * S_L * 4)

// One workgroup per (b, i). 256 threads = 8 wave32s.
// Phase 1: one[l, (j,k)] = sum_m a[b,i,m,l] * a[b,j,k,m]   -> LDS (padded)
// Phase 2: two[j, (k,l)] = sum_m a[b,i,j,m] * a[b,m,k,l]   -> fused KL combine
__global__ __launch_bounds__(256) void associator_loss_main(
    const float* __restrict__ a, float* __restrict__ ws) {
  extern __shared__ float one_lds[];  // 32 * S_L floats

  const int b = blockIdx.x >> 5;
  const int i = blockIdx.x & 31;
  const float* __restrict__ base = a + (size_t)b * 32768;  // a[b] : [32][32][32]
  const float* __restrict__ bi = base + i * 1024;          // a[b,i] : [32][32]

  const int lane = threadIdx.x & 31;
  const int wv = threadIdx.x >> 5;
  const int t15 = lane & 15;
  const int khalf = (lane >> 4) << 1;  // 0 for lanes 0-15, 2 for lanes 16-31
  const int rhalf = (lane >> 4) << 3;  // C/D row offset: 0 or 8

  // ---------------- Phase 1: GEMM "one" into LDS ----------------
  // D[l, n'] with n' = j*32+k ; A1[l,m]=a[b,i,m,l] ; B1[m,n']=a_flat[n'*32+m]
  for (int tile = wv; tile < 128; tile += 8) {
    const int l0 = (tile >> 6) << 4;  // 0 or 16
    const int n0 = (tile & 63) << 4;  // 0..1008
    v8f acc = {};
#pragma unroll
    for (int m0 = 0; m0 < 32; m0 += 4) {
      const int kb = m0 + khalf;
      v2f afrag;
      afrag.x = bi[kb * 32 + (l0 + t15)];        // a[b,i,kb,  l]
      afrag.y = bi[(kb + 1) * 32 + (l0 + t15)];  // a[b,i,kb+1,l]
      v2f bfrag = *(const v2f*)(base + (n0 + t15) * 32 + kb);  // a[b,(j,k),m..m+1]
      acc = __builtin_amdgcn_wmma_f32_16x16x4_f32(
          false, afrag, false, bfrag, (short)0, acc, false, false);
    }
    const int rbase = l0 + rhalf;
    const int col = n0 + t15;              // n' = j*32 + k (j constant per tile)
    const int cj = (col >> 5) * S_J + (col & 31);
#pragma unroll
    for (int v = 0; v < 8; ++v)
      one_lds[(rbase + v) * S_L + cj] = acc[v];
  }
  __syncthreads();

  // ---------------- Phase 2: GEMM "two" + fused KL combine ----------------
  // D[j, n] with n = k*32+l ; A2[j,m]=a[b,i,j,m] ; B2[m,n]=a_flat[m*1024+n]
  float partial = 0.0f;
  for (int tile = wv; tile < 128; tile += 8) {
    const int j0 = (tile >> 6) << 4;
    const int n0 = (tile & 63) << 4;
    v8f acc = {};
#pragma unroll
    for (int m0 = 0; m0 < 32; m0 += 4) {
      const int kb = m0 + khalf;
      v2f afrag = *(const v2f*)(bi + (j0 + t15) * 32 + kb);  // a[b,i,j,m..m+1]
      v2f bfrag;
      bfrag.x = base[kb * 1024 + (n0 + t15)];        // a[b,kb,  (k,l)]
      bfrag.y = base[(kb + 1) * 1024 + (n0 + t15)];  // a[b,kb+1,(k,l)]
      acc = __builtin_amdgcn_wmma_f32_16x16x4_f32(
          false, afrag, false, bfrag, (short)0, acc, false, false);
    }
    const int jrb = j0 + rhalf;
    const int n = n0 + t15;
    const int kk = n >> 5;
    const int ll = n & 31;
    const int obase = ll * S_L + kk;
#pragma unroll
    for (int v = 0; v < 8; ++v) {
      const float t = acc[v];                           // two[b,i,j,k,l]
      const float o = one_lds[obase + (jrb + v) * S_J]; // one[b,i,j,k,l]
      partial += t * (__log2f(t) - __log2f(o));  // scale by ln2 once at the end
    }
  }
  partial *= LN2F;

  // ---------------- Deterministic block reduction ----------------
  __syncthreads();  // done reading one_lds; reuse it
  one_lds[threadIdx.x] = partial;
  __syncthreads();
#pragma unroll
  for (int s = 128; s > 0; s >>= 1) {
    if ((int)threadIdx.x < s) one_lds[threadIdx.x] += one_lds[threadIdx.x + s];
    __syncthreads();
  }
  if (threadIdx.x == 0) ws[blockIdx.x] = one_lds[0];
}

// Single block: sum the 1024 per-workgroup partials, divide by B=32.
__global__ __launch_bounds__(1024) void associator_loss_reduce(
    const float* __restrict__ ws, float* __restrict__ out) {
  __shared__ float s[1024];
  const int t = threadIdx.x;
  s[t] = ws[t];
  __syncthreads();
#pragma unroll
  for (int k = 512; k > 0; k >>= 1) {
    if (t < k) s[t] += s[t + k];
    __syncthreads();
  }
  if (t == 0) out[0] = s[0] * (1.0f / 32.0f);
}

extern "C" void kernel_launch(void* const* d_in, const int* in_sizes, int n_in,
                              void* d_out, int out_size, void* d_ws, size_t ws_size,
                              hipStream_t stream) {
  (void)in_sizes; (void)n_in; (void)out_size; (void)ws_size;
  const float* a = (const float*)d_in[0];  // [32][32][32][32] f32
  float* ws = (float*)d_ws;                // 1024 floats of scratch
  float* out = (float*)d_out;              // scalar f32

  associator_loss_main<<<dim3(1024), dim3(256), LDS_BYTES, stream>>>(a, ws);
  associator_loss_reduce<<<dim3(1), dim3(1024), 0, stream>>>(ws, out);
}